// RSCC_Loss_47012712022644
// MI455X (gfx1250) — compile-verified
//
#include <hip/hip_runtime.h>

#define GDIM 256
#define G3   (GDIM * GDIM * GDIM)
#define NOFF 343
#define RAD  3

typedef __attribute__((ext_vector_type(2))) float v2f;
typedef __attribute__((ext_vector_type(8))) float v8f;

// ---------------------------------------------------------------------------
// Init both min-d2 grids (2 * G3 ints) to sentinel 7 using b128 stores.
// ---------------------------------------------------------------------------
__global__ void init_grids(int* __restrict__ g, long long n4) {
    long long i      = (long long)blockIdx.x * blockDim.x + threadIdx.x;
    long long stride = (long long)gridDim.x * blockDim.x;
    int4* p = (int4*)g;
    const int4 v = make_int4(7, 7, 7, 7);
    for (; i < n4; i += stride) p[i] = v;
}

// ---------------------------------------------------------------------------
// Splat: one thread per (atom, offset) pair; scatter-min of integer d2.
// coords layout is (3, N): [x0..xN-1, y0..yN-1, z0..zN-1].
// Bbox-clamp in the reference reduces to "pos inside [0, G)" per component.
// ---------------------------------------------------------------------------
__global__ void splat_min_d2(const float* __restrict__ coords, int n,
                             const float* __restrict__ ox,
                             const float* __restrict__ oy,
                             const float* __restrict__ oz,
                             const float* __restrict__ vox,
                             int* __restrict__ grid) {
    int tid = blockIdx.x * blockDim.x + threadIdx.x;
    int total = n * NOFF;
    if (tid >= total) return;
    int atom = tid / NOFF;
    int o    = tid - atom * NOFF;
    int o0 = o / 49 - RAD;
    int o1 = (o / 7) % 7 - RAD;
    int o2 = o % 7 - RAD;
    int d2 = o0 * o0 + o1 * o1 + o2 * o2;
    if (d2 > 6) return;                               // sphere mask (81 offsets)

    float v = vox[0];
    int i0 = (int)floorf((coords[atom]         - ox[0]) / v);
    int i1 = (int)floorf((coords[n + atom]     - oy[0]) / v);
    int i2 = (int)floorf((coords[2 * n + atom] - oz[0]) / v);
    if ((unsigned)i0 >= GDIM || (unsigned)i1 >= GDIM || (unsigned)i2 >= GDIM)
        return;                                       // center_ok
    int p0 = i0 + o0, p1 = i1 + o1, p2 = i2 + o2;
    if ((unsigned)p0 >= GDIM || (unsigned)p1 >= GDIM || (unsigned)p2 >= GDIM)
        return;                                       // in_range (clamped bbox)
    atomicMin(&grid[(p0 * GDIM + p1) * GDIM + p2], d2);
}

// ---------------------------------------------------------------------------
// d2 -> exp(-K*d2), K = (pi/3.5)^2 = 0.80568199 ; sentinel 7 -> 0.
// ---------------------------------------------------------------------------
__device__ __forceinline__ float d2prob(int d2) {
    float p = 0.0f;
    p = (d2 == 6) ? 0.00795389f : p;
    p = (d2 == 5) ? 0.01780258f : p;
    p = (d2 == 4) ? 0.03984615f : p;
    p = (d2 == 3) ? 0.08918465f : p;
    p = (d2 == 2) ? 0.19961510f : p;
    p = (d2 == 1) ? 0.44678307f : p;
    p = (d2 == 0) ? 1.0f        : p;
    return p;
}

// ---------------------------------------------------------------------------
// Streaming reduction: acc += s*(s - t) per voxel, b128 loads.
// Wave32 cross-lane reduce via V_WMMA_F32_16X16X4_F32 with B = ones:
//   A vgpr0 = lane partial -> D[m,n] = acc_m + acc_{m+16} (all n equal);
//   sum of the 8 D vgprs + shfl_xor(16) = full wave sum. Deterministic.
// ---------------------------------------------------------------------------
__global__ void reduce_loss(const int* __restrict__ sg, const int* __restrict__ tg,
                            float* __restrict__ partials) {
    const int4* sv = (const int4*)sg;
    const int4* tv = (const int4*)tg;
    const int n4 = G3 / 4;
    int i      = blockIdx.x * blockDim.x + threadIdx.x;
    int stride = gridDim.x * blockDim.x;
    float acc = 0.0f;
    for (; i < n4; i += stride) {
        int4 a = sv[i];
        int4 b = tv[i];
        float s0 = d2prob(a.x), t0 = d2prob(b.x);
        float s1 = d2prob(a.y), t1 = d2prob(b.y);
        float s2 = d2prob(a.z), t2 = d2prob(b.z);
        float s3 = d2prob(a.w), t3 = d2prob(b.w);
        acc = fmaf(s0, s0 - t0, acc);
        acc = fmaf(s1, s1 - t1, acc);
        acc = fmaf(s2, s2 - t2, acc);
        acc = fmaf(s3, s3 - t3, acc);
    }

    // --- wave32 reduction via WMMA (EXEC all-1s: threads reconverged here) ---
    v2f A; A.x = acc;  A.y = 0.0f;
    v2f B; B.x = 1.0f; B.y = 1.0f;
    v8f C = {0.0f, 0.0f, 0.0f, 0.0f, 0.0f, 0.0f, 0.0f, 0.0f};
    C = __builtin_amdgcn_wmma_f32_16x16x4_f32(false, A, false, B,
                                              (short)0, C, false, false);
    float e = C[0] + C[1] + C[2] + C[3] + C[4] + C[5] + C[6] + C[7];
    e += __shfl_xor(e, 16, 32);           // halves hold rows 0-7 / 8-15 sums

    __shared__ float wsum[8];
    int lane = threadIdx.x & 31;
    int wid  = threadIdx.x >> 5;
    if (lane == 0) wsum[wid] = e;
    __syncthreads();
    if (threadIdx.x == 0) {
        float s = 0.0f;
        int nw = blockDim.x >> 5;
        for (int w = 0; w < nw; ++w) s += wsum[w];
        partials[blockIdx.x] = s;
    }
}

// ---------------------------------------------------------------------------
// Final fixed-order sum of block partials -> scalar loss.
// ---------------------------------------------------------------------------
__global__ void final_sum(const float* __restrict__ partials, int nb,
                          float* __restrict__ out) {
    __shared__ float sh[256];
    float s = 0.0f;
    for (int i = threadIdx.x; i < nb; i += blockDim.x) s += partials[i];
    sh[threadIdx.x] = s;
    __syncthreads();
    for (int off = 128; off > 0; off >>= 1) {
        if ((int)threadIdx.x < off) sh[threadIdx.x] += sh[threadIdx.x + off];
        __syncthreads();
    }
    if (threadIdx.x == 0) out[0] = sh[0];
}

// ---------------------------------------------------------------------------
extern "C" void kernel_launch(void* const* d_in, const int* in_sizes, int n_in,
                              void* d_out, int out_size, void* d_ws, size_t ws_size,
                              hipStream_t stream) {
    const float* src = (const float*)d_in[0];   // (1,3,N) flat
    const float* tgt = (const float*)d_in[1];
    const float* x0  = (const float*)d_in[2];
    const float* y0  = (const float*)d_in[3];
    const float* z0  = (const float*)d_in[4];
    const float* vox = (const float*)d_in[5];
    // d_in[6] = grid_dim (== 256, compile-time GDIM)

    const int N = in_sizes[0] / 3;

    int*   sgrid    = (int*)d_ws;
    int*   tgrid    = sgrid + G3;
    float* partials = (float*)(tgrid + G3);

    // 1) init both grids to sentinel d2 = 7
    init_grids<<<2048, 256, 0, stream>>>(sgrid, (long long)(2LL * G3 / 4));

    // 2) scatter-min of d2 for both point clouds
    int nb_splat = (N * NOFF + 255) / 256;
    splat_min_d2<<<nb_splat, 256, 0, stream>>>(src, N, x0, y0, z0, vox, sgrid);
    splat_min_d2<<<nb_splat, 256, 0, stream>>>(tgt, N, x0, y0, z0, vox, tgrid);

    // 3) streaming reduction (bandwidth-bound; grids likely resident in 192MB L2)
    const int RB = 1024;
    reduce_loss<<<RB, 256, 0, stream>>>(sgrid, tgrid, partials);

    // 4) deterministic final sum
    final_sum<<<1, 256, 0, stream>>>(partials, RB, (float*)d_out);
}